// PosEncoding_7619271983661
// MI455X (gfx1250) — compile-verified
//
#include <hip/hip_runtime.h>

// PosEncoding: out[p, 0:3] = c[p, 0:3]
//              out[p, 3 + 6*f + 2*j + {0,1}] = sin/cos(c[p,j] * 2^f * pi)
// Store-bandwidth bound: 504 MB written -> stage per-point rows in LDS, drain
// the contiguous 63KB tile with coalesced b128 (async-from-LDS on gfx1250).

#define AS1 __attribute__((address_space(1)))
#define AS3 __attribute__((address_space(3)))

typedef int v4i __attribute__((ext_vector_type(4)));
typedef AS1 v4i* gchunk_t;   // global-side b128 chunk pointer
typedef AS3 v4i* lchunk_t;   // LDS-side b128 chunk pointer

constexpr int NFREQ = 10;
constexpr int OUTD  = 63;          // 3 + 2*3*10
constexpr int T     = 256;         // threads per block (8 waves of 32)
constexpr int P     = 256;         // points per block

#if __has_builtin(__builtin_amdgcn_global_load_async_to_lds_b128) && \
    __has_builtin(__builtin_amdgcn_global_store_async_from_lds_b128) && \
    __has_builtin(__builtin_amdgcn_s_wait_asynccnt)
#define HAS_ASYNC 1
#else
#define HAS_ASYNC 0
#endif

// v_sin_f32 / v_cos_f32 consume revolutions (radians * 1/(2*pi)).
// sin(c * 2^f * pi) == v_sin(c * 2^(f-1)); c in [0,1) keeps |arg| < 256.
__device__ __forceinline__ float pow2f(int e) {      // 2^e via exponent bits
  return __builtin_bit_cast(float, (127 + e) << 23);
}

__global__ __launch_bounds__(T) void posenc_kernel(const float* __restrict__ in,
                                                   float* __restrict__ out,
                                                   int npts) {
  __shared__ __align__(16) float lin[P * 3];      // 3 KB
  __shared__ __align__(16) float lout[P * OUTD];  // 63 KB
  const int t = threadIdx.x;
  const long long pbase = (long long)blockIdx.x * P;
  const int rem = (int)((long long)npts - pbase);

  if (rem >= P) {
    const float* __restrict__ gin = in + pbase * 3;
    float* __restrict__ gout = out + pbase * (long long)OUTD;

    // ---- Stage A: global -> LDS, coalesced (async path on gfx1250) ----
#if HAS_ASYNC
    if (t < (P * 3) / 4) {  // 192 x b128 chunks
      __builtin_amdgcn_global_load_async_to_lds_b128(
          (gchunk_t)(void*)(gin + t * 4), (lchunk_t)(void*)(lin + t * 4), 0, 0);
    }
    __builtin_amdgcn_s_wait_asynccnt(0);
#else
    for (int i = t; i < P * 3; i += T) lin[i] = gin[i];
#endif
    __syncthreads();

    // ---- Stage B: one point per thread, 63 values into LDS row ----
    // lin reads: word addr 3*t -> gcd(3,64)=1 -> bank-conflict free.
    // lout writes: word stride 63 -> gcd(63,64)=1 -> bank-conflict free.
    const float c0 = lin[t * 3 + 0];
    const float c1 = lin[t * 3 + 1];
    const float c2 = lin[t * 3 + 2];
    float* o = lout + t * OUTD;
    o[0] = c0; o[1] = c1; o[2] = c2;
#pragma unroll
    for (int f = 0; f < NFREQ; ++f) {
      const float rev = pow2f(f - 1);   // (2^f * pi) / (2*pi)
      const float r0 = c0 * rev, r1 = c1 * rev, r2 = c2 * rev;
      o[3 + 6 * f + 0] = __builtin_amdgcn_sinf(r0);
      o[3 + 6 * f + 1] = __builtin_amdgcn_cosf(r0);
      o[3 + 6 * f + 2] = __builtin_amdgcn_sinf(r1);
      o[3 + 6 * f + 3] = __builtin_amdgcn_cosf(r1);
      o[3 + 6 * f + 4] = __builtin_amdgcn_sinf(r2);
      o[3 + 6 * f + 5] = __builtin_amdgcn_cosf(r2);
    }
    __syncthreads();

    // ---- Stage C: drain 63 KB contiguous tile, fully coalesced b128 ----
    constexpr int CHUNKS = (P * OUTD) / 4;  // 4032 float4 chunks
#if HAS_ASYNC
    for (int i = t; i < CHUNKS; i += T) {
      __builtin_amdgcn_global_store_async_from_lds_b128(
          (gchunk_t)(void*)(gout + i * 4), (lchunk_t)(void*)(lout + i * 4), 0, 0);
    }
    __builtin_amdgcn_s_wait_asynccnt(0);
#else
    for (int i = t; i < CHUNKS; i += T) {
      ((float4*)gout)[i] = ((const float4*)lout)[i];
    }
#endif
  } else {
    // ---- Tail block (npts % P != 0): scalar, correctness only ----
    if (t < rem) {
      const long long p = pbase + t;
      const float* ci = in + p * 3;
      const float c0 = ci[0], c1 = ci[1], c2 = ci[2];
      float* o = out + p * (long long)OUTD;
      o[0] = c0; o[1] = c1; o[2] = c2;
#pragma unroll
      for (int f = 0; f < NFREQ; ++f) {
        const float rev = pow2f(f - 1);
        const float r0 = c0 * rev, r1 = c1 * rev, r2 = c2 * rev;
        o[3 + 6 * f + 0] = __builtin_amdgcn_sinf(r0);
        o[3 + 6 * f + 1] = __builtin_amdgcn_cosf(r0);
        o[3 + 6 * f + 2] = __builtin_amdgcn_sinf(r1);
        o[3 + 6 * f + 3] = __builtin_amdgcn_cosf(r1);
        o[3 + 6 * f + 4] = __builtin_amdgcn_sinf(r2);
        o[3 + 6 * f + 5] = __builtin_amdgcn_cosf(r2);
      }
    }
  }
}

extern "C" void kernel_launch(void* const* d_in, const int* in_sizes, int n_in,
                              void* d_out, int out_size, void* d_ws, size_t ws_size,
                              hipStream_t stream) {
  (void)n_in; (void)d_ws; (void)ws_size; (void)out_size;
  const float* coords = (const float*)d_in[0];
  float* out = (float*)d_out;
  const int npts = in_sizes[0] / 3;                 // 2,097,152
  const int blocks = (npts + P - 1) / P;            // 8192
  posenc_kernel<<<blocks, T, 0, stream>>>(coords, out, npts);
}